// PG2Net_90795608637913
// MI455X (gfx1250) — compile-verified
//
#include <hip/hip_runtime.h>
#include <hip/hip_bf16.h>
#include <math.h>

// ---------------- model constants ----------------
#define LOC_SIZE 10000
#define CID_SIZE 300
#define TIM_SIZE 48
#define LOC_EMB 256
#define CID_EMB 64
#define UID_EMB 64
#define TIM_EMB 32
#define HID 500
#define ENC_HID 250
#define SEQ 2048
#define TGT 256
#define HLEN (SEQ - TGT)       // 1792
#define INPUT (LOC_EMB + TIM_EMB + CID_EMB)  // 352

typedef __attribute__((ext_vector_type(2))) float v2f;
typedef __attribute__((ext_vector_type(8))) float v8f;

__device__ __forceinline__ float sigf(float x) { return 1.0f / (1.0f + __expf(-x)); }

template <int NT>
__device__ __forceinline__ float blk_max(float v, float* sm) {
  int tid = threadIdx.x;
  sm[tid] = v; __syncthreads();
#pragma unroll
  for (int s = NT / 2; s > 0; s >>= 1) {
    if (tid < s) sm[tid] = fmaxf(sm[tid], sm[tid + s]);
    __syncthreads();
  }
  float r = sm[0]; __syncthreads();
  return r;
}

template <int NT>
__device__ __forceinline__ float blk_sum(float v, float* sm) {
  int tid = threadIdx.x;
  sm[tid] = v; __syncthreads();
#pragma unroll
  for (int s = NT / 2; s > 0; s >>= 1) {
    if (tid < s) sm[tid] += sm[tid + s];
    __syncthreads();
  }
  float r = sm[0]; __syncthreads();
  return r;
}

// ---------------- WMMA f32 GEMM: C[M,N] = A[M,K] * B + bias ----------------
// BT=true : B is W[N,K] row-major, used transposed (contiguous in K -> b64 loads)
// BT=false: B is plain [K,N] row-major (strided b32 loads)
// One wave computes a 16x32 C slab (two 16x16 tiles sharing the A loads).
// Out-of-range columns are CLAMPED for loads (branchless) and masked on store.
template <bool BT>
__global__ void gemm_wmma_f32(const float* __restrict__ A, const float* __restrict__ B,
                              const float* __restrict__ bias, float* __restrict__ C,
                              int M, int N, int K, int hasBias) {
  const int wave = threadIdx.x >> 5;
  const int lane = threadIdx.x & 31;
  const int tilesN = (N + 15) >> 4;
  const int tilesN2 = (tilesN + 1) >> 1;
  const int tilesM = (M + 15) >> 4;
  const int wid = blockIdx.x * (blockDim.x >> 5) + wave;
  const int tm = wid / tilesN2;
  const int tn0 = (wid - tm * tilesN2) << 1;  // first of two 16-col tiles
  if (tm >= tilesM) return;  // wave-uniform: EXEC stays all-ones below

  const int l16 = lane & 15;
  const int khalf = (lane >> 4) << 1;  // 0 or 2
  const int mA = tm * 16 + l16;        // M is always a multiple of 16 here
  const int n0 = tn0 * 16 + l16;
  const int n1 = n0 + 16;
  const int n0c = min(n0, N - 1);      // branchless clamp: loads always in-bounds
  const int n1c = min(n1, N - 1);

  const float* Arow = A + (long long)mA * K + khalf;
  v8f acc0 = {}, acc1 = {};

  if (BT) {
    const float* B0 = B + (long long)n0c * K + khalf;
    const float* B1 = B + (long long)n1c * K + khalf;
#pragma unroll 4
    for (int k = 0; k < K; k += 4) {
      v2f a  = *(const v2f*)(Arow + k);
      v2f b0 = *(const v2f*)(B0 + k);
      v2f b1 = *(const v2f*)(B1 + k);
      acc0 = __builtin_amdgcn_wmma_f32_16x16x4_f32(false, a, false, b0, (short)0, acc0,
                                                   false, false);
      acc1 = __builtin_amdgcn_wmma_f32_16x16x4_f32(false, a, false, b1, (short)0, acc1,
                                                   false, false);
    }
  } else {
    const float* B0 = B + (long long)khalf * N + n0c;
    const float* B1 = B + (long long)khalf * N + n1c;
#pragma unroll 4
    for (int k = 0; k < K; k += 4) {
      v2f a = *(const v2f*)(Arow + k);
      long long off = (long long)k * N;
      v2f b0, b1;
      b0.x = B0[off]; b0.y = B0[off + N];
      b1.x = B1[off]; b1.y = B1[off + N];
      acc0 = __builtin_amdgcn_wmma_f32_16x16x4_f32(false, a, false, b0, (short)0, acc0,
                                                   false, false);
      acc1 = __builtin_amdgcn_wmma_f32_16x16x4_f32(false, a, false, b1, (short)0, acc1,
                                                   false, false);
    }
  }

  const int mBase = tm * 16 + ((lane >> 4) << 3);  // lanes 0-15: rows v, 16-31: v+8
  if (n0 < N) {
    float bv = hasBias ? bias[n0] : 0.0f;
#pragma unroll
    for (int v = 0; v < 8; ++v) C[(long long)(mBase + v) * N + n0] = acc0[v] + bv;
  }
  if (n1 < N) {
    float bv = hasBias ? bias[n1] : 0.0f;
#pragma unroll
    for (int v = 0; v < 8; ++v) C[(long long)(mBase + v) * N + n1] = acc1[v] + bv;
  }
}

// ---------------- embeddings: x[2048,352] and target_emb ----------------
__global__ void embed_kernel(const int* __restrict__ loc, const int* __restrict__ tim,
                             const int* __restrict__ cid, const int* __restrict__ target,
                             const float* __restrict__ weight, const float* __restrict__ emb_tim_w,
                             const float* __restrict__ weight_cid, float* __restrict__ x,
                             float* __restrict__ temb) {
  int idx = blockIdx.x * blockDim.x + threadIdx.x;
  const int totalX = SEQ * INPUT;      // 720896
  const int totalT = TGT * LOC_EMB;    // 65536
  if (idx < totalX) {
    int t = idx / INPUT, k = idx - t * INPUT;
    float v;
    if (k < LOC_EMB)              v = weight[(long long)loc[t] * LOC_EMB + k];
    else if (k < LOC_EMB+TIM_EMB) v = emb_tim_w[tim[t] * TIM_EMB + (k - LOC_EMB)];
    else                          v = weight_cid[cid[t] * CID_EMB + (k - LOC_EMB - TIM_EMB)];
    x[idx] = v;
  } else if (idx < totalX + totalT) {
    int r = idx - totalX;
    int i = r >> 8, k = r & 255;
    temb[r] = weight[(long long)target[i] * LOC_EMB + k];
  }
}

// ---------------- 3 sequential LSTMs, one persistent block each ----------------
// block 0: fwd encoder, block 1: bwd encoder, block 2: decoder. 1024 threads.
__global__ void lstm3_kernel(const float* __restrict__ Zf, const float* __restrict__ Zb,
                             const float* __restrict__ Zd, const float* __restrict__ Wf,
                             const float* __restrict__ Wb, const float* __restrict__ Wd,
                             float* __restrict__ HH, float* __restrict__ Hdec) {
  __shared__ float h[HID];
  __shared__ float c[HID];
  __shared__ float z[4 * HID];
  const int tid = threadIdx.x;
  const int which = blockIdx.x;

  if (which < 2) {
    const float* Z = (which == 0) ? Zf : Zb;
    const float* W = (which == 0) ? Wf : Wb;
    const int colOff = (which == 0) ? 0 : ENC_HID;
    for (int r = tid; r < ENC_HID; r += 1024) { h[r] = 0.f; c[r] = 0.f; }
    __syncthreads();
    for (int s = 0; s < HLEN; ++s) {
      int row = (which == 0) ? s : (HLEN - 1 - s);
      if (tid < 4 * ENC_HID) {
        const float* w = W + (long long)tid * ENC_HID;
        float acc = Z[(long long)row * (4 * ENC_HID) + tid];
#pragma unroll 5
        for (int j = 0; j < ENC_HID; ++j) acc += w[j] * h[j];
        z[tid] = acc;
      }
      __syncthreads();
      if (tid < ENC_HID) {
        float zi = z[tid], zf = z[ENC_HID + tid], zg = z[2 * ENC_HID + tid],
              zo = z[3 * ENC_HID + tid];
        float cn = sigf(zf) * c[tid] + sigf(zi) * tanhf(zg);
        c[tid] = cn;
        float hn = sigf(zo) * tanhf(cn);
        h[tid] = hn;
        HH[(long long)row * (2 * ENC_HID) + colOff + tid] = hn;
      }
      __syncthreads();
    }
  } else {
    for (int r = tid; r < HID; r += 1024) { h[r] = 0.f; c[r] = 0.f; }
    __syncthreads();
    for (int s = 0; s < TGT; ++s) {
      for (int r = tid; r < 4 * HID; r += 1024) {
        const float* w = Wd + (long long)r * HID;
        float acc = Zd[(long long)s * (4 * HID) + r];
#pragma unroll 5
        for (int j = 0; j < HID; ++j) acc += w[j] * h[j];
        z[r] = acc;
      }
      __syncthreads();
      if (tid < HID) {
        float zi = z[tid], zf = z[HID + tid], zg = z[2 * HID + tid], zo = z[3 * HID + tid];
        float cn = sigf(zf) * c[tid] + sigf(zi) * tanhf(zg);
        c[tid] = cn;
        float hn = sigf(zo) * tanhf(cn);
        h[tid] = hn;
        Hdec[(long long)s * HID + tid] = hn;
      }
      __syncthreads();
    }
  }
}

// ---------------- table attentions: A[256,1792] = sum of 3 softmaxed tables;
//                  S[256,256] = sum of 3 causal-quirk softmaxed tables ------
__global__ void attn_kernel(const int* __restrict__ loc, const int* __restrict__ tim,
                            const int* __restrict__ cid, const float* __restrict__ time_sim,
                            const float* __restrict__ poi_dist,
                            const float* __restrict__ poi_cid_tim, float* __restrict__ Aout,
                            float* __restrict__ Sout) {
  __shared__ float e[HLEN];
  __shared__ float acc[HLEN];
  __shared__ float red[256];
  const int i = blockIdx.x;
  const int tid = threadIdx.x;
  const int* loc_t = loc + HLEN;
  const int* tim_t = tim + HLEN;
  const int* cid_t = cid + HLEN;
  const int li = loc_t[i];
  const int ti = tim_t[i];

  // history attentions
  for (int j = tid; j < HLEN; j += 256) acc[j] = 0.f;
  for (int table = 0; table < 3; ++table) {
    __syncthreads();
    float lmax = -1e30f;
    for (int j = tid; j < HLEN; j += 256) {
      float v;
      if (table == 0)      v = 1.0f / poi_dist[(long long)li * LOC_SIZE + loc[j]];
      else if (table == 1) v = time_sim[ti * TIM_SIZE + tim[j]];
      else                 v = poi_cid_tim[ti * CID_SIZE + cid[j]];
      e[j] = v;
      lmax = fmaxf(lmax, v);
    }
    float mx = blk_max<256>(lmax, red);
    float lsum = 0.f;
    for (int j = tid; j < HLEN; j += 256) {
      float ex = __expf(e[j] - mx);
      e[j] = ex;
      lsum += ex;
    }
    float sum = blk_sum<256>(lsum, red);
    float inv = 1.0f / sum;
    for (int j = tid; j < HLEN; j += 256) acc[j] += e[j] * inv;
  }
  __syncthreads();
  for (int j = tid; j < HLEN; j += 256) Aout[(long long)i * HLEN + j] = acc[j];

  // self attentions (zero-above-diagonal quirk, then full-row softmax)
  __syncthreads();
  for (int j = tid; j < TGT; j += 256) acc[j] = 0.f;
  for (int table = 0; table < 3; ++table) {
    __syncthreads();
    float lmax = -1e30f;
    for (int j = tid; j < TGT; j += 256) {
      float v;
      if (table == 0)      v = 1.0f / poi_dist[(long long)li * LOC_SIZE + loc_t[j]];
      else if (table == 1) v = time_sim[ti * TIM_SIZE + tim_t[j]];
      else                 v = poi_cid_tim[ti * CID_SIZE + cid_t[j]];
      if (j > i) v = 0.0f;
      e[j] = v;
      lmax = fmaxf(lmax, v);
    }
    float mx = blk_max<256>(lmax, red);
    float lsum = 0.f;
    for (int j = tid; j < TGT; j += 256) {
      float ex = __expf(e[j] - mx);
      e[j] = ex;
      lsum += ex;
    }
    float sum = blk_sum<256>(lsum, red);
    float inv = 1.0f / sum;
    for (int j = tid; j < TGT; j += 256) acc[j] += e[j] * inv;
  }
  __syncthreads();
  for (int j = tid; j < TGT; j += 256) Sout[(long long)i * TGT + j] = acc[j];
}

// ---------------- uid embedding -> fc_user ----------------
__global__ void uid_kernel(const int* __restrict__ uid, const float* __restrict__ emb_uid_w,
                           const float* __restrict__ fc_user_w,
                           const float* __restrict__ fc_user_b, float* __restrict__ uvec) {
  __shared__ float eu[UID_EMB];
  int tid = threadIdx.x;
  int u = uid[0];
  if (tid < UID_EMB) eu[tid] = emb_uid_w[(long long)u * UID_EMB + tid];
  __syncthreads();
  if (tid < HID) {
    float acc = fc_user_b[tid];
#pragma unroll
    for (int k = 0; k < UID_EMB; ++k) acc += eu[k] * fc_user_w[tid * UID_EMB + k];
    uvec[tid] = acc;
  }
}

// ---------------- user attention over hidden_history ----------------
__global__ void userattn_kernel(const float* __restrict__ uvec, const float* __restrict__ HH,
                                float* __restrict__ ctxu) {
  __shared__ float w[HLEN];
  __shared__ float red[256];
  __shared__ float uv[HID];
  int tid = threadIdx.x;  // 256
  for (int k = tid; k < HID; k += 256) uv[k] = uvec[k];
  __syncthreads();
  float lmax = -1e30f;
  for (int j = tid; j < HLEN; j += 256) {
    const float* hr = HH + (long long)j * HID;
    float acc = 0.f;
    for (int k = 0; k < HID; ++k) acc += uv[k] * hr[k];
    w[j] = acc;
    lmax = fmaxf(lmax, acc);
  }
  float mx = blk_max<256>(lmax, red);
  float lsum = 0.f;
  for (int j = tid; j < HLEN; j += 256) {
    float ex = __expf(w[j] - mx);
    w[j] = ex;
    lsum += ex;
  }
  float s = blk_sum<256>(lsum, red);
  float inv = 1.0f / s;
  for (int j = tid; j < HLEN; j += 256) w[j] *= inv;
  __syncthreads();
  for (int k = tid; k < HID; k += 256) {
    float acc = 0.f;
    for (int j = 0; j < HLEN; ++j) acc += w[j] * HH[(long long)j * HID + k];
    ctxu[k] = acc;
  }
}

// ---------------- out = [hsatt + uvec, context, ctx_user] ----------------
__global__ void assemble_out(const float* __restrict__ hs, const float* __restrict__ uvec,
                             const float* __restrict__ ctx, const float* __restrict__ ctxu,
                             float* __restrict__ out) {
  int idx = blockIdx.x * blockDim.x + threadIdx.x;
  const int total = TGT * 3 * HID;
  if (idx >= total) return;
  int i = idx / (3 * HID), k = idx - i * (3 * HID);
  float v;
  if (k < HID)          v = hs[i * HID + k] + uvec[k];
  else if (k < 2 * HID) v = ctx[i * HID + (k - HID)];
  else                  v = ctxu[k - 2 * HID];
  out[idx] = v;
}

// ---------------- in-place row log-softmax over y[256,10000] ----------------
__global__ void logsoftmax_kernel(float* __restrict__ y) {
  __shared__ float red[512];
  const int i = blockIdx.x;
  const int tid = threadIdx.x;
  float* row = y + (long long)i * LOC_SIZE;
  float lmax = -1e30f;
  for (int j = tid; j < LOC_SIZE; j += 512) lmax = fmaxf(lmax, row[j]);
  float mx = blk_max<512>(lmax, red);
  float lsum = 0.f;
  for (int j = tid; j < LOC_SIZE; j += 512) lsum += __expf(row[j] - mx);
  float s = blk_sum<512>(lsum, red);
  float lse = mx + logf(s);
  for (int j = tid; j < LOC_SIZE; j += 512) row[j] = row[j] - lse;
}

// ---------------- host side ----------------
static inline void gemm(hipStream_t s, const float* A, const float* B, const float* bias,
                        float* C, int M, int N, int K, bool bT, int hasBias) {
  int tilesM = (M + 15) / 16, tilesN = (N + 15) / 16;
  int tilesN2 = (tilesN + 1) / 2;
  int tiles = tilesM * tilesN2;
  int blocks = (tiles + 7) / 8;  // 8 waves (32t) per 256-thread block
  if (bT)
    gemm_wmma_f32<true><<<blocks, 256, 0, s>>>(A, B, bias, C, M, N, K, hasBias);
  else
    gemm_wmma_f32<false><<<blocks, 256, 0, s>>>(A, B, bias, C, M, N, K, hasBias);
}

extern "C" void kernel_launch(void* const* d_in, const int* in_sizes, int n_in,
                              void* d_out, int out_size, void* d_ws, size_t ws_size,
                              hipStream_t stream) {
  const int* loc    = (const int*)d_in[0];
  const int* tim    = (const int*)d_in[1];
  const int* cid    = (const int*)d_in[2];
  const int* uid    = (const int*)d_in[3];
  const int* target = (const int*)d_in[4];
  // d_in[5] = target_len (compile-time TGT here)
  const float* time_sim   = (const float*)d_in[6];
  const float* poi_dist   = (const float*)d_in[7];
  const float* poi_cid_t  = (const float*)d_in[8];
  const float* weight     = (const float*)d_in[9];
  const float* weight_cid = (const float*)d_in[10];
  const float* emb_uid_w  = (const float*)d_in[11];
  const float* emb_tim_w  = (const float*)d_in[12];
  const float* enc_Wih_f  = (const float*)d_in[13];
  const float* enc_Whh_f  = (const float*)d_in[14];
  const float* enc_b_f    = (const float*)d_in[15];
  const float* enc_Wih_b  = (const float*)d_in[16];
  const float* enc_Whh_b  = (const float*)d_in[17];
  const float* enc_b_b    = (const float*)d_in[18];
  const float* dec_Wih    = (const float*)d_in[19];
  const float* dec_Whh    = (const float*)d_in[20];
  const float* dec_b      = (const float*)d_in[21];
  const float* fc_user_w  = (const float*)d_in[22];
  const float* fc_user_b  = (const float*)d_in[23];
  const float* fc_final_w = (const float*)d_in[24];
  const float* fc_final_b = (const float*)d_in[25];
  const float* fc_final2_w = (const float*)d_in[26];
  const float* fc_final2_b = (const float*)d_in[27];

  float* out = (float*)d_out;                          // [0, 2560000): score (y in place)
  float* y1  = out + (long long)TGT * LOC_SIZE;        // next 128000: y1
  float* temb = y1 + (long long)TGT * HID;             // next 65536: target_emb

  // workspace layout (floats)
  float* ws = (float*)d_ws;
  float* x     = ws;                                      // 2048*352
  float* Zf    = x     + (long long)SEQ * INPUT;          // 1792*1000
  float* Zb    = Zf    + (long long)HLEN * 4 * ENC_HID;   // 1792*1000
  float* Zd    = Zb    + (long long)HLEN * 4 * ENC_HID;   // 256*2000
  float* HH    = Zd    + (long long)TGT * 4 * HID;        // 1792*500
  float* Hdec  = HH    + (long long)HLEN * 2 * ENC_HID;   // 256*500
  float* uvec  = Hdec  + (long long)TGT * HID;            // 500
  float* ctxu  = uvec  + 512;                             // 500
  float* Aattn = ctxu  + 512;                             // 256*1792
  float* Ssum  = Aattn + (long long)TGT * HLEN;           // 256*256
  float* ctx   = Ssum  + (long long)TGT * TGT;            // 256*500
  float* hsatt = ctx   + (long long)TGT * HID;            // 256*500
  float* outb  = hsatt + (long long)TGT * HID;            // 256*1500

  // 1) embeddings + target_emb
  {
    int total = SEQ * INPUT + TGT * LOC_EMB;
    embed_kernel<<<(total + 255) / 256, 256, 0, stream>>>(loc, tim, cid, target, weight,
                                                          emb_tim_w, weight_cid, x, temb);
  }

  // 2) LSTM input projections (WMMA): Z = X @ Wih^T + b
  gemm(stream, x, enc_Wih_f, enc_b_f, Zf, HLEN, 4 * ENC_HID, INPUT, true, 1);
  gemm(stream, x, enc_Wih_b, enc_b_b, Zb, HLEN, 4 * ENC_HID, INPUT, true, 1);
  gemm(stream, x + (long long)HLEN * INPUT, dec_Wih, dec_b, Zd, TGT, 4 * HID, INPUT, true, 1);

  // 3) recurrences (3 persistent blocks)
  lstm3_kernel<<<3, 1024, 0, stream>>>(Zf, Zb, Zd, enc_Whh_f, enc_Whh_b, dec_Whh, HH, Hdec);

  // 4) table attentions (independent of LSTM)
  attn_kernel<<<TGT, 256, 0, stream>>>(loc, tim, cid, time_sim, poi_dist, poi_cid_t, Aattn, Ssum);

  // 5) uid vector + user attention over hidden_history
  uid_kernel<<<1, 512, 0, stream>>>(uid, emb_uid_w, fc_user_w, fc_user_b, uvec);
  userattn_kernel<<<1, 256, 0, stream>>>(uvec, HH, ctxu);

  // 6) context = Aattn @ HH ; hsatt = Ssum @ Hdec   (WMMA, plain-B addressing)
  gemm(stream, Aattn, HH, nullptr, ctx, TGT, HID, HLEN, false, 0);
  gemm(stream, Ssum, Hdec, nullptr, hsatt, TGT, HID, TGT, false, 0);

  // 7) assemble out = [hsatt + uvec, ctx, ctxu]
  {
    int total = TGT * 3 * HID;
    assemble_out<<<(total + 255) / 256, 256, 0, stream>>>(hsatt, uvec, ctx, ctxu, outb);
  }

  // 8) heads (WMMA): y -> d_out (then in-place log-softmax), y1 -> d_out tail
  gemm(stream, outb, fc_final_w, fc_final_b, out, TGT, LOC_SIZE, 3 * HID, true, 1);
  gemm(stream, outb, fc_final2_w, fc_final2_b, y1, TGT, HID, 3 * HID, true, 1);

  // 9) log-softmax rows of y in place
  logsoftmax_kernel<<<TGT, 512, 0, stream>>>(out);
}